// RRNNforGRUCell_32641751449644
// MI455X (gfx1250) — compile-verified
//
#include <hip/hip_runtime.h>

// ---------------------------------------------------------------------------
// RRNN tree-structure search for MI455X (gfx1250).
// Key facts exploited:
//  * forward pass == vectors selected during search -> only run the search.
//  * per round only ONE new candidate needs SL/SR matvecs (cached otherwise):
//    80 total 4096x4096 f32 matvecs, ~5.1 GB HBM traffic -> bandwidth bound.
//  * all data-dependent bookkeeping kept on-device (graph-capture safe).
// Matvecs use V_WMMA_F32_16X16X4_F32 (16-wide output tile, K-step 4).
// ---------------------------------------------------------------------------

#define HDIM 4096
#define MULTF 0.001f
#define NROUNDS 9
#define NSLOTS 12          // x, h, 0, G0..G8
#define MAXC 8             // 3 + max 5 comps
#define MAXP 28            // C*(C-1)/2 for C=8
#define NCOMBO (MAXP * 4 * 2)      // pair * k * binop = 224 blocks
#define NSCORE (NCOMBO * 4)        // * activation = 896 scores

typedef __attribute__((ext_vector_type(2))) float v2f;
typedef __attribute__((ext_vector_type(8))) float v8f;

// ---------------------------------------------------------------------------
// Matvec: out[k][:] = v @ M[k] for 4 L matrices and 4 R matrices.
// grid = 8*256 single-wave blocks; each wave computes a 16-wide output tile.
// WMMA mapping (D = A x B + C, 16x16x4 f32):
//   A[m][kk] = M[h+kk][d+m]   (transposed tile; lanes 0-15 hold K=0,1 in
//                              VGPR0,1; lanes 16-31 hold K=2,3)
//   B[kk][n] = v[h+kk] broadcast over all n  -> every D column = result
//   D[m][n]  = sum_kk M[h+kk][d+m]*v[h+kk]   -> y[d+m]
// ---------------------------------------------------------------------------
__global__ __launch_bounds__(32) void matvec_wmma_kernel(
    const float* __restrict__ v, const float* __restrict__ Lm,
    const float* __restrict__ Rm, float* __restrict__ outSL,
    float* __restrict__ outSR) {
  const int matId = blockIdx.x >> 8;   // 0..3 -> L, 4..7 -> R
  const int tile  = blockIdx.x & 255;
  const int d     = tile << 4;
  const float* M;
  float* outv;
  if (matId < 4) {
    M = Lm + (size_t)matId * HDIM * HDIM;
    outv = outSL + matId * HDIM;
  } else {
    M = Rm + (size_t)(matId - 4) * HDIM * HDIM;
    outv = outSR + (matId - 4) * HDIM;
  }
  const int lane = threadIdx.x;
  const int half = lane >> 4;      // 0: K=0,1   1: K=2,3
  const int l16  = lane & 15;      // output row m within tile
  const float* p0 = M + (size_t)(2 * half) * HDIM + d + l16;
  const float* vb = v + 2 * half;

  v8f acc0 = {}, acc1 = {}, acc2 = {}, acc3 = {};
#define WMMA_STEP(ACC, HH)                                                   \
  {                                                                          \
    v2f A, B;                                                                \
    A[0] = p0[(size_t)(HH) * HDIM];                                          \
    A[1] = p0[(size_t)(HH) * HDIM + HDIM];                                   \
    B[0] = vb[(HH)];                                                         \
    B[1] = vb[(HH) + 1];                                                     \
    ACC = __builtin_amdgcn_wmma_f32_16x16x4_f32(false, A, false, B,          \
                                                (short)0, ACC, false, false);\
  }
  for (int h = 0; h < HDIM; h += 16) {
    WMMA_STEP(acc0, h);
    WMMA_STEP(acc1, h + 4);
    WMMA_STEP(acc2, h + 8);
    WMMA_STEP(acc3, h + 12);
  }
#undef WMMA_STEP
  v8f acc = (acc0 + acc1) + (acc2 + acc3);
  // D layout: VGPR g, lanes 0-15 -> M=g ; lanes 16-31 -> M=g+8 (any N column)
  if (lane == 0) {
#pragma unroll
    for (int g = 0; g < 8; g++) outv[d + g] = acc[g];
  } else if (lane == 16) {
#pragma unroll
    for (int g = 0; g < 8; g++) outv[d + 8 + g] = acc[g];
  }
}

// ---------------------------------------------------------------------------
// Init: zero the "0"-candidate slot's SL/SR, reset comps state, sum(w).
// ---------------------------------------------------------------------------
__global__ __launch_bounds__(256) void init_kernel(float* __restrict__ SL,
                                                   float* __restrict__ SR,
                                                   const float* __restrict__ w,
                                                   float* __restrict__ sumw,
                                                   int* __restrict__ state) {
  const int tid = threadIdx.x;
  for (int t = tid; t < 4 * HDIM; t += 256) {
    SL[(size_t)2 * 4 * HDIM + t] = 0.0f;   // slot 2 = zero vector
    SR[(size_t)2 * 4 * HDIM + t] = 0.0f;
  }
  float s = 0.0f;
  for (int t = tid; t < HDIM; t += 256) s += w[t];
  __shared__ float red[256];
  red[tid] = s;
  __syncthreads();
  for (int st = 128; st > 0; st >>= 1) {
    if (tid < st) red[tid] += red[tid + st];
    __syncthreads();
  }
  if (tid == 0) {
    sumw[0] = red[0];
    state[0] = 0;   // comps count
  }
}

// ---------------------------------------------------------------------------
// Scoring: one block per (pair p, k, binop) combo; writes 4 masked scores.
// ---------------------------------------------------------------------------
__global__ __launch_bounds__(256) void score_kernel(
    const float* __restrict__ SL, const float* __restrict__ SR,
    const float* __restrict__ b, const float* __restrict__ w,
    const float* __restrict__ sumw, const int* __restrict__ state,
    float* __restrict__ scoreBuf, int r) {
  const int tid  = threadIdx.x;
  const int cIdx = blockIdx.x;           // p*8 + k*2 + bi
  const int p  = cIdx >> 3;
  const int k  = (cIdx >> 1) & 3;
  const int bi = cIdx & 1;               // 0 = mul, 1 = add
  const float NEG = -__builtin_inff();

  const int c = state[0];
  const int C = 3 + c;
  const int P = C * (C - 1) / 2;
  if (p >= P) {
    if (tid < 4) scoreBuf[cIdx * 4 + tid] = NEG;
    return;
  }
  // decode p -> (i,j) in np.triu_indices order
  int i = 0, rp = p;
  while (rp >= C - 1 - i) { rp -= C - 1 - i; i++; }
  const int j = i + 1 + rp;
  // pair validity per remaining-iterations bookkeeping
  const int rem = NROUNDS - r;
  bool valid;
  if (rem > c)        valid = true;
  else if (rem == c)  valid = (j >= 3);
  else                valid = (i >= 3);   // rem == c-1
  if (!valid) {
    if (tid < 4) scoreBuf[cIdx * 4 + tid] = NEG;
    return;
  }
  const int si = (i < 3) ? i : state[1 + (i - 3)];
  const int sj = (j < 3) ? j : state[1 + (j - 3)];
  const float* slp = SL + ((size_t)si * 4 + k) * HDIM;
  const float* srp = SR + ((size_t)sj * 4 + k) * HDIM;
  const float* bp  = b + (size_t)k * HDIM;

  float s_sig = 0.0f, s_tanh = 0.0f, s_lin = 0.0f, m = 0.0f;
  for (int t = tid; t < HDIM; t += 256) {
    const float sl = slp[t], sr = srp[t];
    const float res = (bi == 0 ? sl * sr : sl + sr) + bp[t];
    const float wt = w[t];
    s_sig  += wt / (1.0f + expf(-res));
    s_tanh += wt * tanhf(res);
    s_lin  += wt * res;
    m = fmaxf(m, fabsf(res));
  }
  __shared__ float r0[256], r1[256], r2[256], r3[256];
  r0[tid] = s_sig; r1[tid] = s_tanh; r2[tid] = s_lin; r3[tid] = m;
  __syncthreads();
  for (int st = 128; st > 0; st >>= 1) {
    if (tid < st) {
      r0[tid] += r0[tid + st];
      r1[tid] += r1[tid + st];
      r2[tid] += r2[tid + st];
      r3[tid] = fmaxf(r3[tid], r3[tid + st]);
    }
    __syncthreads();
  }
  if (tid == 0) {
    const bool big = (r3[0] >= 1.0f);
    scoreBuf[cIdx * 4 + 0] = MULTF * r0[0];                              // sigmoid
    scoreBuf[cIdx * 4 + 1] = MULTF * r1[0];                              // tanh
    scoreBuf[cIdx * 4 + 2] = big ? NEG : MULTF * (sumw[0] - r2[0]);      // minus
    scoreBuf[cIdx * 4 + 3] = big ? NEG : MULTF * r2[0];                  // identity
  }
}

// ---------------------------------------------------------------------------
// Select: argmax over 896 scores (numpy row-major tie-break), emit G[r],
// score[r], h_next (r==8), and update comps bookkeeping.
// ---------------------------------------------------------------------------
__global__ __launch_bounds__(256) void select_kernel(
    const float* __restrict__ SL, const float* __restrict__ SR,
    const float* __restrict__ b, const float* __restrict__ scoreBuf,
    int* __restrict__ state, float* __restrict__ Gout,
    float* __restrict__ hnext, float* __restrict__ scoresOut, int r) {
  const int tid = threadIdx.x;
  float bestV = -__builtin_inff();
  int bestI = 0x7fffffff;
  for (int t = tid; t < NSCORE; t += 256) {
    const float v = scoreBuf[t];
    if (v > bestV) { bestV = v; bestI = t; }   // ascending stride: first-max
  }
  __shared__ float sv[256];
  __shared__ int   sx[256];
  sv[tid] = bestV; sx[tid] = bestI;
  __syncthreads();
  for (int st = 128; st > 0; st >>= 1) {
    if (tid < st) {
      if (sv[tid + st] > sv[tid] ||
          (sv[tid + st] == sv[tid] && sx[tid + st] < sx[tid])) {
        sv[tid] = sv[tid + st];
        sx[tid] = sx[tid + st];
      }
    }
    __syncthreads();
  }
  __shared__ int sel_i, sel_j, sel_k, sel_bi, sel_ai, sel_si, sel_sj;
  if (tid == 0) {
    const int flat = sx[0];                 // p*32 + k*8 + bi*4 + ai
    const int ai = flat & 3;
    const int bi = (flat >> 2) & 1;
    const int k  = (flat >> 3) & 3;
    const int p  = flat >> 5;
    const int c = state[0];
    const int C = 3 + c;
    int i = 0, rp = p;
    while (rp >= C - 1 - i) { rp -= C - 1 - i; i++; }
    const int j = i + 1 + rp;
    sel_i = i; sel_j = j; sel_k = k; sel_bi = bi; sel_ai = ai;
    sel_si = (i < 3) ? i : state[1 + (i - 3)];
    sel_sj = (j < 3) ? j : state[1 + (j - 3)];
    scoresOut[r] = sv[0];
  }
  __syncthreads();
  const float* slp = SL + ((size_t)sel_si * 4 + sel_k) * HDIM;
  const float* srp = SR + ((size_t)sel_sj * 4 + sel_k) * HDIM;
  const float* bp  = b + (size_t)sel_k * HDIM;
  for (int t = tid; t < HDIM; t += 256) {
    const float res =
        (sel_bi == 0 ? slp[t] * srp[t] : slp[t] + srp[t]) + bp[t];
    float val;
    if (sel_ai == 0)      val = MULTF / (1.0f + expf(-res));
    else if (sel_ai == 1) val = MULTF * tanhf(res);
    else if (sel_ai == 2) val = MULTF * (1.0f - res);
    else                  val = MULTF * res;
    Gout[(size_t)r * HDIM + t] = val;
    if (r == NROUNDS - 1) hnext[t] = val;
  }
  if (tid == 0) {
    const int c = state[0];
    const int newSlot = 3 + r;              // slot where G[r]'s SL/SR will go
    const int i = sel_i, j = sel_j;
    if (i > 2 && j > 2) {                   // merge two comps
      state[1 + (i - 3)] = newSlot;
      for (int t = j - 3; t < c - 1; t++) state[1 + t] = state[1 + t + 1];
      state[0] = c - 1;
    } else if (j > 2) {                     // replace comps[j-3]
      state[1 + (j - 3)] = newSlot;
    } else {                                // append
      state[1 + c] = newSlot;
      state[0] = c + 1;
    }
  }
}

// ---------------------------------------------------------------------------
extern "C" void kernel_launch(void* const* d_in, const int* in_sizes, int n_in,
                              void* d_out, int out_size, void* d_ws,
                              size_t ws_size, hipStream_t stream) {
  (void)in_sizes; (void)n_in; (void)out_size; (void)ws_size;
  const float* x = (const float*)d_in[0];
  const float* h = (const float*)d_in[1];
  const float* L = (const float*)d_in[2];
  const float* R = (const float*)d_in[3];
  const float* b = (const float*)d_in[4];
  const float* w = (const float*)d_in[5];

  float* out       = (float*)d_out;
  float* hnext     = out;                     // [H]
  float* Gout      = out + HDIM;              // [9, H]
  float* scoresOut = out + 10 * HDIM;         // [9]

  // workspace layout (floats): SL[12][4][H] | SR[12][4][H] | scores | sumw | state
  float* SL       = (float*)d_ws;
  float* SR       = SL + (size_t)NSLOTS * 4 * HDIM;
  float* scoreBuf = SR + (size_t)NSLOTS * 4 * HDIM;
  float* sumw     = scoreBuf + NSCORE;
  int*   state    = (int*)(sumw + 1);

  init_kernel<<<1, 256, 0, stream>>>(SL, SR, w, sumw, state);
  // slot 0 = x, slot 1 = h_prev (slot 2 = zeros, handled by init)
  matvec_wmma_kernel<<<2048, 32, 0, stream>>>(x, L, R, SL + (size_t)0 * 4 * HDIM,
                                              SR + (size_t)0 * 4 * HDIM);
  matvec_wmma_kernel<<<2048, 32, 0, stream>>>(h, L, R, SL + (size_t)1 * 4 * HDIM,
                                              SR + (size_t)1 * 4 * HDIM);
  for (int r = 0; r < NROUNDS; r++) {
    score_kernel<<<NCOMBO, 256, 0, stream>>>(SL, SR, b, w, sumw, state,
                                             scoreBuf, r);
    select_kernel<<<1, 256, 0, stream>>>(SL, SR, b, scoreBuf, state, Gout,
                                         hnext, scoresOut, r);
    if (r < NROUNDS - 1) {
      // G[r] becomes candidate slot 3+r next round
      matvec_wmma_kernel<<<2048, 32, 0, stream>>>(
          Gout + (size_t)r * HDIM, L, R, SL + (size_t)(3 + r) * 4 * HDIM,
          SR + (size_t)(3 + r) * 4 * HDIM);
    }
  }
}